// EmbeddingLayer_10591389352095
// MI455X (gfx1250) — compile-verified
//
#include <hip/hip_runtime.h>

typedef __attribute__((ext_vector_type(2))) float v2f;
typedef __attribute__((ext_vector_type(8))) float v8f;

#define SLOPE 0.01f
#define BATCH 8
#define SEQ   512
#define DIM   64
#define TILE_M 64
#define KC     32          // K chunk (double buffered)
#define NCH    (SEQ / KC)  // 16 chunks
#define PCH    36          // chunk LDS pitch: 36*lane mod 64 -> 16 distinct banks

__device__ __forceinline__ float leakyf(float x) { return x >= 0.f ? x : SLOPE * x; }
// -leaky(-x): identity for x<=0, slope*x for x>0
__device__ __forceinline__ float qleakyf(float x) { return x <= 0.f ? x : SLOPE * x; }

// CDNA5 async global->LDS DMA (ASYNCcnt tracked), 16B per lane, no VGPR bounce.
__device__ __forceinline__ void async_ld_b128(unsigned lds_addr, const void* gptr) {
    asm volatile("global_load_async_to_lds_b128 %0, %1, off"
                 :: "v"(lds_addr), "v"((unsigned long long)(uintptr_t)gptr)
                 : "memory");
}
#define WAIT_ASYNCCNT0() asm volatile("s_wait_asynccnt 0x0" ::: "memory")

__global__ __launch_bounds__(128) void gnn_embed_fused(
    const float* __restrict__ prev,  // (B,N,D)
    const float* __restrict__ adj,   // (B,N,N)
    const float* __restrict__ nf,    // (B,N,4)
    const float* __restrict__ ef,    // (B,N,N)
    const float* __restrict__ W1,    // (D,4)
    const float* __restrict__ W2,    // (D,D)
    const float* __restrict__ W3,    // (D,D)
    const float* __restrict__ W4,    // (D,1)
    float* __restrict__ out)         // (B,N,D)
{
    __shared__ float  adjA[2][64][PCH];   // adj K-chunk, row-major [m][k], async-filled
    __shared__ float  prevT[2][64][PCH];  // prev K-chunk, TRANSPOSED [n][k]
    __shared__ float  TL[64][68];         // T tile between GEMM1 and GEMM2
    __shared__ float  SpL[64], SqL[64], nrmL[64];
    __shared__ float  CpL[64], CqL[64];
    __shared__ float4 w1L[64];
    __shared__ float4 nfL[64];
    __shared__ float  w4L[64];

    const int tid   = threadIdx.x;
    const int wave  = tid >> 5;      // wave32 (gfx1250)
    const int lane  = tid & 31;
    const int half  = lane >> 4;     // K-pair select for f32 WMMA fragments
    const int lm    = lane & 15;
    const int b     = blockIdx.y;
    const int row0g = blockIdx.x * TILE_M;
    const int wrow0 = wave * 16;

    const float* adjB  = adj  + ((size_t)b * SEQ + row0g) * SEQ;  // tile base
    const float* prevB = prev + (size_t)b * SEQ * DIM;

    const unsigned ldsA0 = (unsigned)(uintptr_t)&adjA[0][0][0];
    const unsigned ldsA1 = (unsigned)(uintptr_t)&adjA[1][0][0];

    // ---- prologue: launch async DMA for adj chunk 0, pipeline prev regs -----
    {
        #pragma unroll
        for (int j = 0; j < 4; ++j) {
            int idx = tid + j * 128;
            int rr = idx >> 3, c4 = idx & 7;                 // 64 rows x 8 xfloat4
            async_ld_b128(ldsA0 + (unsigned)(rr * PCH + c4 * 4) * 4u,
                          adjB + (size_t)rr * SEQ + c4 * 4);
        }
    }
    float4 pr[4];
    #pragma unroll
    for (int j = 0; j < 4; ++j) {                            // prev chunk 0
        int idx = tid + j * 128;
        int kk = idx >> 4, c4 = idx & 15;
        pr[j] = *(const float4*)&prevB[(size_t)kk * DIM + c4 * 4];
    }
    #pragma unroll
    for (int j = 0; j < 4; ++j) {                            // store transposed
        int idx = tid + j * 128;
        int kk = idx >> 4, c4 = idx & 15;
        prevT[0][c4 * 4 + 0][kk] = pr[j].x;
        prevT[0][c4 * 4 + 1][kk] = pr[j].y;
        prevT[0][c4 * 4 + 2][kk] = pr[j].z;
        prevT[0][c4 * 4 + 3][kk] = pr[j].w;
    }
    #pragma unroll
    for (int j = 0; j < 4; ++j) {                            // prev chunk 1
        int idx = tid + j * 128;
        int kk = idx >> 4, c4 = idx & 15;
        pr[j] = *(const float4*)&prevB[(size_t)(KC + kk) * DIM + c4 * 4];
    }

    // ---- (a) small operands (overlaps with in-flight DMA) -------------------
    if (tid < 64) {
        w4L[tid] = W4[tid];
        w1L[tid] = ((const float4*)W1)[tid];
        nfL[tid] = ((const float4*)nf)[(size_t)b * SEQ + row0g + tid];
        __builtin_prefetch(W2 + tid * DIM, 0, 0);            // global_prefetch_b8
        __builtin_prefetch(W3 + tid * DIM, 0, 0);
    }

    // ---- (b) per-row reductions: norm, Sp, Sq over neighbors j --------------
    for (int rr = 0; rr < 16; ++rr) {
        const int    row  = wrow0 + rr;
        const size_t base = ((size_t)b * SEQ + row0g + row) * SEQ;
        const float4* a4 = (const float4*)(adj + base);
        const float4* e4 = (const float4*)(ef + base);
        float sn = 0.f, sp = 0.f, sq = 0.f;
        for (int j = lane; j < SEQ / 4; j += 32) {
            float4 av = a4[j], ev = e4[j];
            sn += av.x + av.y + av.z + av.w;
            sp += av.x * leakyf(ev.x) + av.y * leakyf(ev.y)
                + av.z * leakyf(ev.z) + av.w * leakyf(ev.w);
            sq += av.x * qleakyf(ev.x) + av.y * qleakyf(ev.y)
                + av.z * qleakyf(ev.z) + av.w * qleakyf(ev.w);
        }
        #pragma unroll
        for (int off = 16; off; off >>= 1) {
            sn += __shfl_xor(sn, off, 32);
            sp += __shfl_xor(sp, off, 32);
            sq += __shfl_xor(sq, off, 32);
        }
        if (lane == 0) { nrmL[row] = sn; SpL[row] = sp; SqL[row] = sq; }
    }
    __syncthreads();   // w4L visible for (c)

    // ---- (c) rank-2 collapse of x3: Cp[e], Cq[e] ----------------------------
    if (tid < 64) {
        float cp = 0.f, cq = 0.f;
        const float* w3r = W3 + tid * DIM;
        for (int d = 0; d < DIM; ++d) {
            float w = w4L[d];
            float v = w3r[d] * w;
            if (w >= 0.f) cp += v; else cq += v;
        }
        CpL[tid] = cp; CqL[tid] = cq;
    }

    // ---- (d) GEMM1: T = adj_tile(64x512) @ prev(512x64), fp32 WMMA ----------
    // Double-buffered pipeline, one barrier per chunk:
    //   wait(c) -> barrier (buf^1 now free in ALL waves) -> issue c+1 -> compute c
    const v8f vzero = {0.f, 0.f, 0.f, 0.f, 0.f, 0.f, 0.f, 0.f};
    v8f acc[4] = {vzero, vzero, vzero, vzero};

    for (int c = 0; c < NCH; ++c) {
        WAIT_ASYNCCNT0();          // adj chunk c landed in LDS
        __syncthreads();           // visible; every wave done reading buf^1
        const int buf = c & 1;
        if (c + 1 < NCH) {
            const unsigned ldsN = buf ? ldsA0 : ldsA1;
            const int kc1 = (c + 1) * KC;
            #pragma unroll
            for (int j = 0; j < 4; ++j) {   // async DMA chunk c+1 (flies over compute c)
                int idx = tid + j * 128;
                int rr = idx >> 3, c4 = idx & 7;
                async_ld_b128(ldsN + (unsigned)(rr * PCH + c4 * 4) * 4u,
                              adjB + (size_t)rr * SEQ + kc1 + c4 * 4);
            }
            #pragma unroll
            for (int j = 0; j < 4; ++j) {   // stage prev chunk c+1 (regs->LDS, transposed)
                int idx = tid + j * 128;
                int kk = idx >> 4, c4 = idx & 15;
                prevT[buf ^ 1][c4 * 4 + 0][kk] = pr[j].x;
                prevT[buf ^ 1][c4 * 4 + 1][kk] = pr[j].y;
                prevT[buf ^ 1][c4 * 4 + 2][kk] = pr[j].z;
                prevT[buf ^ 1][c4 * 4 + 3][kk] = pr[j].w;
            }
            if (c + 2 < NCH) {
                const int kc2 = (c + 2) * KC;
                #pragma unroll
                for (int j = 0; j < 4; ++j) {
                    int idx = tid + j * 128;
                    int kk = idx >> 4, c4 = idx & 15;
                    pr[j] = *(const float4*)&prevB[(size_t)(kc2 + kk) * DIM + c4 * 4];
                }
            }
        }
        #pragma unroll
        for (int k = 0; k < KC; k += 4) {
            // A fragment (16x4 f32): lane = m + 16*(k>=2), vgpr = k&1 -> ds_load_b64
            v2f a = *(const v2f*)&adjA[buf][wrow0 + lm][k + 2 * half];
            #pragma unroll
            for (int nb = 0; nb < 4; ++nb) {
                // B fragment (4x16 f32): transposed LDS -> contiguous ds_load_b64
                v2f bf = *(const v2f*)&prevT[buf][nb * 16 + lm][k + 2 * half];
                acc[nb] = __builtin_amdgcn_wmma_f32_16x16x4_f32(
                    false, a, false, bf, (short)0, acc[nb], false, false);
            }
        }
    }

    // ---- (e) spill T slab (each wave writes/reads only its own 16 rows) -----
    #pragma unroll
    for (int nb = 0; nb < 4; ++nb)
        #pragma unroll
        for (int r = 0; r < 8; ++r)
            TL[wrow0 + r + 8 * half][nb * 16 + lm] = acc[nb][r];

    // ---- (f) GEMM2: x2 = T(64x64) @ W2^T(64x64), fp32 WMMA ------------------
    v8f acc2[4] = {vzero, vzero, vzero, vzero};
    #pragma unroll
    for (int k = 0; k < DIM; k += 4) {
        v2f a;
        a.x = TL[wrow0 + lm][k + 2 * half];
        a.y = TL[wrow0 + lm][k + 2 * half + 1];
        #pragma unroll
        for (int nb = 0; nb < 4; ++nb) {
            // B[k][n] = W2[n][k] : row-major W2 -> aligned global 8B load
            const float* w2p = W2 + (size_t)(nb * 16 + lm) * DIM + k + 2 * half;
            v2f bf; bf.x = w2p[0]; bf.y = w2p[1];
            acc2[nb] = __builtin_amdgcn_wmma_f32_16x16x4_f32(
                false, a, false, bf, (short)0, acc2[nb], false, false);
        }
    }

    // ---- (g) epilogue: out = leaky(x1 + x2 + x3) ----------------------------
    #pragma unroll
    for (int nb = 0; nb < 4; ++nb) {
        #pragma unroll
        for (int r = 0; r < 8; ++r) {
            int row = wrow0 + r + 8 * half;   // tile-local row (C/D layout)
            int e   = nb * 16 + lm;
            float nrm = nrmL[row];
            nrm = (nrm == 0.f) ? 1.f : nrm;
            float x3v = (SpL[row] * CpL[e] + SqL[row] * CqL[e]) / nrm;
            float4 n4 = nfL[row], w14 = w1L[e];
            float x1v = n4.x * w14.x + n4.y * w14.y + n4.z * w14.z + n4.w * w14.w;
            float v = x1v + acc2[nb][r] + x3v;
            out[((size_t)b * SEQ + row0g + row) * DIM + e] = leakyf(v);
        }
    }
}

extern "C" void kernel_launch(void* const* d_in, const int* in_sizes, int n_in,
                              void* d_out, int out_size, void* d_ws, size_t ws_size,
                              hipStream_t stream) {
    (void)in_sizes; (void)n_in; (void)out_size; (void)d_ws; (void)ws_size;
    const float* prev = (const float*)d_in[0];
    const float* adj  = (const float*)d_in[1];
    const float* nf   = (const float*)d_in[2];
    const float* ef   = (const float*)d_in[3];
    const float* W1   = (const float*)d_in[4];
    const float* W2   = (const float*)d_in[5];
    const float* W3   = (const float*)d_in[6];
    const float* W4   = (const float*)d_in[7];
    float* out = (float*)d_out;

    dim3 grid(SEQ / TILE_M, BATCH);   // 8 row-tiles x 8 batches
    gnn_embed_fused<<<grid, 128, 0, stream>>>(prev, adj, nf, ef, W1, W2, W3, W4, out);
}